// voting_layer_86002425135160
// MI455X (gfx1250) — compile-verified
//
#include <hip/hip_runtime.h>
#include <hip/hip_bf16.h>
#include <stdint.h>

typedef __attribute__((ext_vector_type(2))) float v2f;
typedef __attribute__((ext_vector_type(8))) float v8f;

#define BATCH   4
#define NTOK    32768
#define EDIM    64
#define NKEEP   19660            // int(0.6 * 32768)
#define TOTTOK  (BATCH * NTOK)   // 131072

// ---------------------------------------------------------------------------
// Kernel 1: fold all linear layers into a single 256-element weight vector.
// w_eff[h] = sum_o hr_w1[h,o]*hr_w2[o]
// t[j]     = sum_m nv_w2[j,m]*nv_w3[m]
// wv[i]    = sum_j nv_w1[i,j]*t[j]
// w_comb[k*4+h] = wv[k] * w_eff[h]
// ---------------------------------------------------------------------------
__global__ void prep_weights(const float* __restrict__ hr_w1,
                             const float* __restrict__ hr_w2,
                             const float* __restrict__ nv_w1,
                             const float* __restrict__ nv_w2,
                             const float* __restrict__ nv_w3,
                             float* __restrict__ wc) {
    __shared__ float t[64];
    __shared__ float weff[4];
    const int tid = threadIdx.x;            // 64 threads
    float acc = 0.f;
    for (int m = 0; m < 8; ++m) acc += nv_w2[tid * 8 + m] * nv_w3[m];
    t[tid] = acc;
    if (tid < 4) {
        float we = 0.f;
        for (int o = 0; o < 4; ++o) we += hr_w1[tid * 4 + o] * hr_w2[o];
        weff[tid] = we;
    }
    __syncthreads();
    float wv = 0.f;
    for (int j = 0; j < 64; ++j) wv += nv_w1[tid * 64 + j] * t[j];
    for (int h = 0; h < 4; ++h) wc[tid * 4 + h] = wv * weff[h];
}

// ---------------------------------------------------------------------------
// Kernel 2: fused score GEMV via V_WMMA_F32_16X16X4_F32.
// One wave -> 16 tokens. A = w_comb slice replicated over 16 rows,
// B = 16 token rows (K slice of 4). 64 accumulating WMMAs cover K=256.
// Emits sortable 64-bit keys: (~sortable(score) << 32) | token_in_batch,
// so an ascending sort == descending score with ascending-index tie-break
// (matches jax.lax.top_k ordering).
// ---------------------------------------------------------------------------
__global__ __launch_bounds__(256) void score_wmma(
        const float* __restrict__ att,
        const float* __restrict__ wc,
        unsigned long long* __restrict__ keys) {
    const int lane = threadIdx.x & 31;                 // wave32
    const int wave = threadIdx.x >> 5;                 // 8 waves / block
    const int tile = blockIdx.x * 8 + wave;            // 8192 tiles of 16 tokens
    const int tok  = tile * 16 + (lane & 15);
    const int koff = (lane >> 4) << 1;                 // lanes 0-15 -> K{0,1}, 16-31 -> K{2,3}

    const float* __restrict__ arow = att + (size_t)tok * 256 + koff;
    const float* __restrict__ wrow = wc + koff;

    v8f acc = {0.f, 0.f, 0.f, 0.f, 0.f, 0.f, 0.f, 0.f};
#pragma unroll 4
    for (int s = 0; s < 64; ++s) {
        v2f a, b;
        a.x = wrow[4 * s];
        a.y = wrow[4 * s + 1];
        b.x = arow[4 * s];
        b.y = arow[4 * s + 1];
        // D = A(16x4,f32) * B(4x16,f32) + C(16x16,f32)
        acc = __builtin_amdgcn_wmma_f32_16x16x4_f32(
                false, a, false, b, (short)0, acc, false, false);
    }
    // C layout: VGPR0, lanes 0-15 hold row M=0 cols N=0..15 (all rows identical)
    if (lane < 16) {
        const float score = acc[0];
        unsigned u = __float_as_uint(score);
        unsigned srt = (u & 0x80000000u) ? ~u : (u | 0x80000000u);  // ascending-sortable
        unsigned dsc = ~srt;                                        // descending-sortable
        keys[tok] = ((unsigned long long)dsc << 32) | (unsigned)(tok & (NTOK - 1));
    }
}

// ---------------------------------------------------------------------------
// Kernel 3: per-batch bitonic sort of 32768 64-bit keys in LDS (256 KB of the
// 320 KB WGP LDS), then emit the first NKEEP indices (descending score).
// ---------------------------------------------------------------------------
__global__ __launch_bounds__(1024) void sort_topk(
        const unsigned long long* __restrict__ keys,
        int* __restrict__ idx) {
    extern __shared__ unsigned long long lkeys[];      // 32768 * 8B = 256 KB
    const int b   = blockIdx.x;
    const int tid = threadIdx.x;

    for (int i = tid; i < NTOK; i += 1024)
        lkeys[i] = keys[b * NTOK + i];
    __syncthreads();

    for (int k = 2; k <= NTOK; k <<= 1) {
        for (int j = k >> 1; j > 0; j >>= 1) {
            for (int i = tid; i < NTOK; i += 1024) {
                const int p = i ^ j;
                if (p > i) {
                    const bool asc = ((i & k) == 0);
                    unsigned long long a = lkeys[i];
                    unsigned long long c = lkeys[p];
                    if ((a > c) == asc) { lkeys[i] = c; lkeys[p] = a; }
                }
            }
            __syncthreads();
        }
    }

    for (int i = tid; i < NKEEP; i += 1024)
        idx[b * NKEEP + i] = (int)(lkeys[i] & 0xFFFFFFFFull);
}

// ---------------------------------------------------------------------------
// Kernel 4: gather kept tokens of x [b, NKEEP, 64] and coords [b, 2, NKEEP, 1].
// ---------------------------------------------------------------------------
__global__ void gather_out(const float* __restrict__ x,
                           const float* __restrict__ coords,
                           const int* __restrict__ idx,
                           float* __restrict__ xout,
                           float* __restrict__ cout) {
    const int stride = gridDim.x * blockDim.x;
    const int total_x = BATCH * NKEEP * EDIM;
    for (int g = blockIdx.x * blockDim.x + threadIdx.x; g < total_x; g += stride) {
        const int c = g & (EDIM - 1);
        const int r = g >> 6;                 // b*NKEEP + k
        const int b = r / NKEEP;
        const int k = r - b * NKEEP;
        const int src = idx[b * NKEEP + k];
        xout[g] = x[((size_t)b * NTOK + src) * EDIM + c];
    }
    const int total_c = BATCH * 2 * NKEEP;
    for (int g = blockIdx.x * blockDim.x + threadIdx.x; g < total_c; g += stride) {
        const int k  = g % NKEEP;
        const int t  = g / NKEEP;             // b*2 + ch
        const int b  = t >> 1;
        const int ch = t & 1;
        const int src = idx[b * NKEEP + k];
        cout[g] = coords[((size_t)(b * 2 + ch)) * NTOK + src];
    }
}

// ---------------------------------------------------------------------------
extern "C" void kernel_launch(void* const* d_in, const int* in_sizes, int n_in,
                              void* d_out, int out_size, void* d_ws, size_t ws_size,
                              hipStream_t stream) {
    (void)in_sizes; (void)n_in; (void)out_size; (void)ws_size;

    const float* x      = (const float*)d_in[0];   // [4, 32768, 64]
    const float* att_nh = (const float*)d_in[1];   // [131072, 4, 8, 8]
    const float* coords = (const float*)d_in[2];   // [4, 2, 32768, 1]
    const float* hr_w1  = (const float*)d_in[3];
    const float* hr_w2  = (const float*)d_in[5];
    const float* nv_w1  = (const float*)d_in[7];
    const float* nv_w2  = (const float*)d_in[9];
    const float* nv_w3  = (const float*)d_in[11];
    // biases (d_in[4],[6],[8],[10],[12]) are token-independent -> irrelevant to top-k

    // workspace layout
    char* ws = (char*)d_ws;
    float* wc                 = (float*)ws;                        // 256 floats
    unsigned long long* keys  = (unsigned long long*)(ws + 1024);  // 131072 * 8 B
    int* idx                  = (int*)(ws + 1024 + (size_t)TOTTOK * 8); // 4*19660 ints

    float* xout = (float*)d_out;                      // 4*19660*64
    float* cout = xout + (size_t)BATCH * NKEEP * EDIM; // 4*2*19660

    prep_weights<<<1, 64, 0, stream>>>(hr_w1, hr_w2, nv_w1, nv_w2, nv_w3, wc);

    // 131072 tokens / 16 per wave / 8 waves per block = 1024 blocks
    score_wmma<<<1024, 256, 0, stream>>>(att_nh, wc, keys);

    sort_topk<<<BATCH, 1024, NTOK * sizeof(unsigned long long), stream>>>(keys, idx);

    gather_out<<<512, 256, 0, stream>>>(x, coords, idx, xout, cout);
}